// GradConvSDF_2559800508971
// MI455X (gfx1250) — compile-verified
//
#include <hip/hip_runtime.h>
#include <stdint.h>

// Problem constants (fixed by the reference)
#define NDIMS   3
#define DIL     0.01f
#define MAXD    1.0f
#define GDIM    64
#define G3      (GDIM*GDIM*GDIM)
#define NSDF    8
#define BATCH   8
#define NPTS    131072
#define MOBJ    4
#define INV2DIL 50.0f   // 1/(2*DIL)

// LDS staging layout: 28 pose dwords + 4 scales + 4 idxs(bit-cast) + 8 sdf_sizes = 44 dwords
#define LDS_POSE   0
#define LDS_SCALE  28
#define LDS_IDX    32
#define LDS_SIZE   36
#define LDS_TOT    44

__global__ __launch_bounds__(256) void convsdf_grad_kernel(
    const float* __restrict__ locs,      // [B,N,3]
    const int*   __restrict__ idxs,      // [B,M]
    const float* __restrict__ poses,     // [B,M,7] (t.xyz, q.wxyz)
    const float* __restrict__ scales,    // [B,M]
    const float* __restrict__ sdfs,      // [NSDF,G,G,G]
    const float* __restrict__ sdf_sizes, // [NSDF]
    float*       __restrict__ out)       // [B,N,3]
{
    __shared__ float sraw[LDS_TOT];

    const int tid = threadIdx.x;
    const int b   = blockIdx.y;

    // ---- Stage per-batch object data into LDS: ONE predicated CDNA5 async
    // issue, per-lane 64-bit VGPR address (no saddr -> no branch cascade).
    if (tid < LDS_TOT) {
        const unsigned t = (unsigned)tid;
        uint64_t ga;
        if (t < 28)      ga = (uint64_t)(uintptr_t)(poses  + (size_t)b * (MOBJ * 7)) + 4ull * t;
        else if (t < 32) ga = (uint64_t)(uintptr_t)(scales + (size_t)b * MOBJ)       + 4ull * (t - 28);
        else if (t < 36) ga = (uint64_t)(uintptr_t)(idxs   + (size_t)b * MOBJ)       + 4ull * (t - 32);
        else             ga = (uint64_t)(uintptr_t)(sdf_sizes)                        + 4ull * (t - 36);
        const unsigned lds_addr = (unsigned)(uintptr_t)(&sraw[0]) + 4u * t;
        asm volatile("global_load_async_to_lds_b32 %0, %1, off"
                     :
                     : "v"(lds_addr), "v"(ga)
                     : "memory");
    }
    asm volatile("s_wait_asynccnt 0" ::: "memory");
    __syncthreads();

    const int n = blockIdx.x * blockDim.x + tid;
    if (n >= NPTS) return;

    const size_t pbase = ((size_t)b * NPTS + (size_t)n) * 3;

    // 12-byte vector load of this particle's position (-> global_load_b96).
    float p3[3];
    __builtin_memcpy(p3, locs + pbase, 12);
    const float px = p3[0], py = p3[1], pz = p3[2];

    // Per-axis min accumulators for the -DIL and +DIL taps.
    // Init to MAXD implements both the invalid->MAXD mask and the final min(.,MAXD) clamp.
    float accm0 = MAXD, accm1 = MAXD, accm2 = MAXD;
    float accp0 = MAXD, accp1 = MAXD, accp2 = MAXD;

    #pragma unroll
    for (int m = 0; m < MOBJ; ++m) {
        const float tx = sraw[LDS_POSE + m * 7 + 0];
        const float ty = sraw[LDS_POSE + m * 7 + 1];
        const float tz = sraw[LDS_POSE + m * 7 + 2];
        float qw = sraw[LDS_POSE + m * 7 + 3];
        float qx = sraw[LDS_POSE + m * 7 + 4];
        float qy = sraw[LDS_POSE + m * 7 + 5];
        float qz = sraw[LDS_POSE + m * 7 + 6];
        const float rn = rsqrtf(qw * qw + qx * qx + qy * qy + qz * qz);
        qw *= rn; qx *= rn; qy *= rn; qz *= rn;

        // Rotate-by-conjugate matrix Rc = R(q)^T; local = Rc * (p - t)
        const float r00 = 1.f - 2.f * (qy * qy + qz * qz);
        const float r01 = 2.f * (qx * qy + qw * qz);
        const float r02 = 2.f * (qx * qz - qw * qy);
        const float r10 = 2.f * (qx * qy - qw * qz);
        const float r11 = 1.f - 2.f * (qx * qx + qz * qz);
        const float r12 = 2.f * (qy * qz + qw * qx);
        const float r20 = 2.f * (qx * qz + qw * qy);
        const float r21 = 2.f * (qy * qz - qw * qx);
        const float r22 = 1.f - 2.f * (qx * qx + qy * qy);

        const float sc     = sraw[LDS_SCALE + m];
        const int   oi     = __float_as_int(sraw[LDS_IDX + m]);
        const float cell   = sc * sraw[LDS_SIZE + oi];
        const float invcel = 1.0f / cell;
        const int   baseI  = oi * G3;

        // Base sample point in grid units: u = Rc*(p - t) / cell
        const float dx = px - tx, dy = py - ty, dz = pz - tz;
        const float ux = (r00 * dx + r01 * dy + r02 * dz) * invcel;
        const float uy = (r10 * dx + r11 * dy + r12 * dz) * invcel;
        const float uz = (r20 * dx + r21 * dy + r22 * dz) * invcel;

        // Per-axis step in grid units: (DIL/cell) * column d of Rc
        const float sD = DIL * invcel;
        const float stx[3] = { sD * r00, sD * r01, sD * r02 };
        const float sty[3] = { sD * r10, sD * r11, sD * r12 };
        const float stz[3] = { sD * r20, sD * r21, sD * r22 };

        #pragma unroll
        for (int d = 0; d < NDIMS; ++d) {
            #pragma unroll
            for (int s = 0; s < 2; ++s) {
                const float vx = (s == 0) ? (ux - stx[d]) : (ux + stx[d]);
                const float vy = (s == 0) ? (uy - sty[d]) : (uy + sty[d]);
                const float vz = (s == 0) ? (uz - stz[d]) : (uz + stz[d]);
                const float fx = floorf(vx), fy = floorf(vy), fz = floorf(vz);
                // validity via min3/max3: all three floors in [0,64)
                const float fmn = fminf(fminf(fx, fy), fz);
                const float fmx = fmaxf(fmaxf(fx, fy), fz);
                const bool valid = (fmn >= 0.0f) & (fmx < 64.0f);
                int gx = (int)fx; gx = gx < 0 ? 0 : (gx > 63 ? 63 : gx);
                int gy = (int)fy; gy = gy < 0 ? 0 : (gy > 63 ? 63 : gy);
                int gz = (int)fz; gz = gz < 0 ? 0 : (gz > 63 ? 63 : gz);
                const int lin = (gx * GDIM + gy) * GDIM + gz;
                // Unconditional gather at the clamped index (matches take-then-where),
                // so all 24 gathers can issue as one clause and hide L2 latency.
                const float sv  = sdfs[baseI + lin];
                const float val = valid ? sv * sc : MAXD;
                if (s == 0) {
                    if (d == 0) accm0 = fminf(accm0, val);
                    if (d == 1) accm1 = fminf(accm1, val);
                    if (d == 2) accm2 = fminf(accm2, val);
                } else {
                    if (d == 0) accp0 = fminf(accp0, val);
                    if (d == 1) accp1 = fminf(accp1, val);
                    if (d == 2) accp2 = fminf(accp2, val);
                }
            }
        }
    }

    // 12-byte vector store of the gradient (-> global_store_b96).
    float o3[3];
    o3[0] = (accp0 - accm0) * INV2DIL;
    o3[1] = (accp1 - accm1) * INV2DIL;
    o3[2] = (accp2 - accm2) * INV2DIL;
    __builtin_memcpy(out + pbase, o3, 12);
}

extern "C" void kernel_launch(void* const* d_in, const int* in_sizes, int n_in,
                              void* d_out, int out_size, void* d_ws, size_t ws_size,
                              hipStream_t stream) {
    const float* locs      = (const float*)d_in[0];
    const int*   idxs      = (const int*)  d_in[1];
    const float* poses     = (const float*)d_in[2];
    const float* scales    = (const float*)d_in[3];
    const float* sdfs      = (const float*)d_in[4];
    const float* sdf_sizes = (const float*)d_in[5];
    float*       out       = (float*)d_out;

    dim3 grid(NPTS / 256, BATCH, 1);
    dim3 block(256, 1, 1);
    convsdf_grad_kernel<<<grid, block, 0, stream>>>(locs, idxs, poses, scales,
                                                    sdfs, sdf_sizes, out);
}